// PLASoftmax_30159260352944
// MI455X (gfx1250) — compile-verified
//
#include <hip/hip_runtime.h>
#include <hip/hip_bf16.h>

#define ROW_LEN     512
#define WAVES_PB    8
#define ROWS_PW     4
#define THREADS_PB  256

// ---------------- CDNA5 async global->LDS path (guarded) ----------------
#if defined(__HIP_DEVICE_COMPILE__) && __has_builtin(__builtin_amdgcn_global_load_async_to_lds_b128)
#define HAVE_ASYNC 1
#else
#define HAVE_ASYNC 0
#endif

#if defined(__HIP_DEVICE_COMPILE__)
# if __has_builtin(__builtin_amdgcn_s_wait_asynccnt)
#  define WAIT_ASYNC(n) __builtin_amdgcn_s_wait_asynccnt(n)
# else
#  define WAIT_ASYNC(n) asm volatile("s_wait_asynccnt " #n ::: "memory")
# endif
#else
# define WAIT_ASYNC(n)
#endif

#if HAVE_ASYNC
typedef int v4i_cd5 __attribute__((ext_vector_type(4)));
typedef __attribute__((address_space(1))) v4i_cd5* gptr_v4i;
typedef __attribute__((address_space(3))) v4i_cd5* lptr_v4i;
#endif

// Copy one 2KB row (512 f32) global -> LDS: 4 x (wave32 * b128) chunks.
__device__ __forceinline__ void copy_row_async(const float* __restrict__ g, float* l, int lane) {
#if HAVE_ASYNC
#pragma unroll
  for (int j = 0; j < 4; ++j) {
    const char* gp = (const char*)g + (j * 512 + lane * 16);
    char*       lp = (char*)l + (j * 512 + lane * 16);
    __builtin_amdgcn_global_load_async_to_lds_b128((gptr_v4i)gp, (lptr_v4i)lp, 0, 0);
  }
#else
#pragma unroll
  for (int j = 0; j < 4; ++j) {
    *(float4*)((char*)l + j * 512 + lane * 16) =
        *(const float4*)((const char*)g + j * 512 + lane * 16);
  }
#endif
}

__device__ __forceinline__ float wave_max(float v) {
#pragma unroll
  for (int off = 16; off > 0; off >>= 1) v = fmaxf(v, __shfl_xor(v, off, 32));
  return v;
}
__device__ __forceinline__ float wave_sum(float v) {
#pragma unroll
  for (int off = 16; off > 0; off >>= 1) v += __shfl_xor(v, off, 32);
  return v;
}

__global__ void pla_ws_init(unsigned int* p) { p[0] = 0u; }

// Pass 1: PLA softmax per row + global abs-max reduction.
// One wave per row, ROWS_PW rows per wave, double-buffered async staging in LDS.
__global__ __launch_bounds__(THREADS_PB)
void pla_softmax_pass1(const float* __restrict__ scores,
                       const float* __restrict__ cm,
                       const float* __restrict__ cc,
                       const float* __restrict__ iv,
                       float* __restrict__ out,
                       unsigned int* __restrict__ absmax_bits,
                       int nrows) {
  __shared__ float s_m[12];
  __shared__ float s_c[12];
  __shared__ float s_iv[12];
  __shared__ float s_buf[WAVES_PB][2][ROW_LEN];

  const int tid = threadIdx.x;
  if (tid < 12)       s_m[tid]       = cm[tid];
  else if (tid < 24)  s_c[tid - 12]  = cc[tid - 12];
  else if (tid < 36)  s_iv[tid - 24] = iv[tid - 24];
  __syncthreads();

  const int lane = tid & 31;
  const int wave = tid >> 5;
  const long long row0 = ((long long)blockIdx.x * WAVES_PB + wave) * ROWS_PW;

  float tmax = 0.0f;

  if (row0 < nrows)
    copy_row_async(scores + row0 * ROW_LEN, s_buf[wave][0], lane);

#pragma unroll
  for (int r = 0; r < ROWS_PW; ++r) {
    const long long row = row0 + r;
    const bool active = (row < nrows);
    const bool next_ok = (r + 1 < ROWS_PW) && (row + 1 < nrows);

    if (next_ok) {
      copy_row_async(scores + (row + 1) * ROW_LEN, s_buf[wave][(r + 1) & 1], lane);
      WAIT_ASYNC(4);   // current row's 4 copies done; prefetch stays in flight
    } else {
      WAIT_ASYNC(0);
    }
    if (!active) continue;

    const float* lb = s_buf[wave][r & 1];

    float xv[16];
#pragma unroll
    for (int j = 0; j < 4; ++j) {
      float4 v = *(const float4*)(lb + j * 128 + lane * 4);
      xv[j * 4 + 0] = v.x; xv[j * 4 + 1] = v.y;
      xv[j * 4 + 2] = v.z; xv[j * 4 + 3] = v.w;
    }

    // 1) row max (wave butterfly over 16 elems/lane)
    float mx = -3.402823466e+38f;
#pragma unroll
    for (int k = 0; k < 16; ++k) mx = fmaxf(mx, xv[k]);
    mx = wave_max(mx);

    // 2-3) Q32.26 snap + PLA exp
    float ev[16];
    float psum = 0.0f;
#pragma unroll
    for (int k = 0; k < 16; ++k) {
      float sh = xv[k] - mx;
      float t  = rintf(sh * 67108864.0f);            // * 2^26, round-half-even
      t = fmaxf(t, -2147483648.0f);                  // f32(-2^31)
      t = fminf(t,  2147483648.0f);                  // f32(2^31-1) == 2^31
      float fx = t * 1.4901161193847656e-08f;        // * 2^-26 (exact)
      float xc = fminf(fmaxf(fx, -10.0f), 0.0f);
      int idx = 0;
#pragma unroll
      for (int i = 1; i < 12; ++i) idx += (xc >= s_iv[i]) ? 1 : 0;  // searchsorted right - 1
      idx = idx > 10 ? 10 : idx;
      float e = s_m[idx] * xc + s_c[idx];
      ev[k] = e;
      psum += e;
    }

    // 4) normalize
    psum = wave_sum(psum);
    const float inv = 1.0f / (psum + 1e-9f);

    float* orow = out + row * ROW_LEN;
#pragma unroll
    for (int j = 0; j < 4; ++j) {
      float4 o;
      o.x = ev[j * 4 + 0] * inv; o.y = ev[j * 4 + 1] * inv;
      o.z = ev[j * 4 + 2] * inv; o.w = ev[j * 4 + 3] * inv;
      tmax = fmaxf(tmax, fmaxf(fmaxf(fabsf(o.x), fabsf(o.y)),
                               fmaxf(fabsf(o.z), fabsf(o.w))));
      *(float4*)(orow + j * 128 + lane * 4) = o;
    }
  }

  // tensor-wide abs-max: uint compare == float compare for non-negative values
  tmax = wave_max(tmax);
  if (lane == 0) atomicMax(absmax_bits, __float_as_uint(tmax));
}

// Pass 2: per-tensor symmetric INT8 fake-quant (forward value == q).
__global__ __launch_bounds__(256)
void pla_quant_pass2(float* __restrict__ out,
                     const unsigned int* __restrict__ absmax_bits,
                     int n4) {
  const int i = blockIdx.x * blockDim.x + threadIdx.x;
  if (i >= n4) return;
  const float absmax = __uint_as_float(absmax_bits[0]);
  const float scale  = fmaxf(absmax / 127.0f, 1e-8f);
  const float inv    = 1.0f / scale;

  float4 v = ((const float4*)out)[i];
  v.x = fminf(fmaxf(rintf(v.x * inv), -127.0f), 127.0f) * scale;
  v.y = fminf(fmaxf(rintf(v.y * inv), -127.0f), 127.0f) * scale;
  v.z = fminf(fmaxf(rintf(v.z * inv), -127.0f), 127.0f) * scale;
  v.w = fminf(fmaxf(rintf(v.w * inv), -127.0f), 127.0f) * scale;
  ((float4*)out)[i] = v;
}

extern "C" void kernel_launch(void* const* d_in, const int* in_sizes, int n_in,
                              void* d_out, int out_size, void* d_ws, size_t ws_size,
                              hipStream_t stream) {
  const float* scores = (const float*)d_in[0];
  const float* cm     = (const float*)d_in[1];
  const float* cc     = (const float*)d_in[2];
  const float* iv     = (const float*)d_in[3];
  float* out          = (float*)d_out;
  unsigned int* ws    = (unsigned int*)d_ws;

  const int ntot  = in_sizes[0];           // 25,165,824
  const int nrows = ntot / ROW_LEN;        // 49,152

  hipLaunchKernelGGL(pla_ws_init, dim3(1), dim3(1), 0, stream, ws);

  const int rows_per_block = WAVES_PB * ROWS_PW;  // 32
  const int nblocks = (nrows + rows_per_block - 1) / rows_per_block;  // 1536
  hipLaunchKernelGGL(pla_softmax_pass1, dim3(nblocks), dim3(THREADS_PB), 0, stream,
                     scores, cm, cc, iv, out, ws, nrows);

  const int n4 = ntot / 4;                 // 6,291,456
  hipLaunchKernelGGL(pla_quant_pass2, dim3((n4 + 255) / 256), dim3(256), 0, stream,
                     out, ws, n4);
}